// Spatial_Weighting_29240137351352
// MI455X (gfx1250) — compile-verified
//
#include <hip/hip_runtime.h>

// Spectral weighting: per (b,t,h) problem (384 total):
//   feats = K K^T  (fp32 WMMA 16x16x4), W = feats>0.2 ? 1 : 1e-5 (stored as bitmask)
//   S = D^-1/2 W D^-1/2, Fiedler vector of M = I - S via deflated power iteration
//   on B = I + S (u0 = D^{1/2}1/||.|| deflated analytically), then sign-invariant
//   bipartition + softmax. One workgroup per problem, all LDS-resident.

#define NPROB 384
#define N     196
#define D     64
#define NT    13            // 13x13 tiles of 16x16 cover 208x208
#define KSTR  65            // padded LDS row stride for K (floats); gcd(65,64)=1 -> conflict free
#define MW    8             // mask words per row (256 bits >= 208)
#define THREADS 256
#define NITER 350
#define THRESH 0.2f
#define EPSW  1e-5f

typedef __attribute__((ext_vector_type(2))) float v2f;
typedef __attribute__((ext_vector_type(8))) float v8f;

__device__ __forceinline__ float block_sum(float v, float* red, int tid) {
    __syncthreads();                 // protect previous use of red (WAR)
    red[tid] = v;
    __syncthreads();
    for (int off = 128; off > 0; off >>= 1) {
        if (tid < off) red[tid] += red[tid + off];
        __syncthreads();
    }
    return red[0];
}

__device__ __forceinline__ float block_max(float v, float* red, int tid) {
    __syncthreads();
    red[tid] = v;
    __syncthreads();
    for (int off = 128; off > 0; off >>= 1) {
        if (tid < off) red[tid] = fmaxf(red[tid], red[tid + off]);
        __syncthreads();
    }
    return red[0];
}

__global__ __launch_bounds__(THREADS)
void spatial_weighting_kernel(const float* __restrict__ qkv, float* __restrict__ out) {
    __shared__ unsigned int maskS[N * MW];       // 6272 B : W bitmask (1 bit per edge)
    __shared__ float Kl[208 * KSTR];             // 54080 B: K tile; reused as vectors in phase 2

    // phase-2 vector overlays on Kl (K no longer needed after WMMA phase)
    float* u0   = Kl;                // 256 floats
    float* invs = Kl + 256;
    float* zv   = Kl + 512;
    float* red  = Kl + 768;          // 256-slot reduction scratch

    const int tid  = threadIdx.x;
    const int lane = tid & 31;
    const int wave = tid >> 5;
    const int p    = blockIdx.x;     // problem id: ((b*8+t)*12+h), matches output order

    const float* Kg = qkv + (size_t)1 * NPROB * N * D + (size_t)p * N * D;   // qkv[1][p]

    // ---- Phase 0: init masks, stage K into LDS (pad rows 196..207 with 0) ----
    for (int i = tid; i < N * MW; i += THREADS) maskS[i] = 0u;
    for (int i = tid; i < N * D; i += THREADS) {
        int r = i >> 6, c = i & 63;
        Kl[r * KSTR + c] = Kg[i];
    }
    for (int i = tid; i < (208 - N) * D; i += THREADS) {
        int r = i >> 6, c = i & 63;
        Kl[(N + r) * KSTR + c] = 0.0f;
    }
    __syncthreads();

    // ---- Phase 1: feats = K K^T via V_WMMA_F32_16X16X4_F32, threshold -> bitmask ----
    const int m16   = lane & 15;
    const int khalf = lane >> 4;     // lanes 16-31 carry K = +2 (A) per ISA layout
    for (int tile = wave; tile < NT * NT; tile += (THREADS / 32)) {
        const int I = tile / NT, J = tile % NT;
        const int rowA = (I * 16 + m16) * KSTR;
        const int rowB = (J * 16 + m16) * KSTR;   // B = K^T tile: same per-lane indexing
        v8f acc = {0.f, 0.f, 0.f, 0.f, 0.f, 0.f, 0.f, 0.f};
        #pragma unroll
        for (int kc = 0; kc < D / 4; ++kc) {
            const int kb = kc * 4 + khalf * 2;
            v2f a, b;
            a.x = Kl[rowA + kb];     a.y = Kl[rowA + kb + 1];
            b.x = Kl[rowB + kb];     b.y = Kl[rowB + kb + 1];
            acc = __builtin_amdgcn_wmma_f32_16x16x4_f32(
                      false, a, false, b, (short)0, acc, false, false);
        }
        // C layout: vreg r -> (M=r, N=lane) lanes 0-15 ; (M=r+8, N=lane-16) lanes 16-31
        const bool colok = (J * 16 + m16) < N;
        const int  sh    = (J & 1) * 16;
        const int  wrd   = J >> 1;
        #pragma unroll
        for (int r = 0; r < 8; ++r) {
            unsigned int bal = __builtin_amdgcn_ballot_w32((acc[r] > THRESH) && colok);
            if (lane == 0) {
                int i0 = I * 16 + r;
                if (i0 < N) atomicOr(&maskS[i0 * MW + wrd], (bal & 0xFFFFu) << sh);
                int i1 = I * 16 + r + 8;
                if (i1 < N) atomicOr(&maskS[i1 * MW + wrd], ((bal >> 16) & 0xFFFFu) << sh);
            }
        }
    }
    __syncthreads();

    // ---- Phase 2a: degrees d_i = #set + (N - #set)*eps ; u0 = D^{1/2}1 / ||.|| ----
    float d_i = 0.f, isq = 0.f;
    if (tid < N) {
        int nset = 0;
        #pragma unroll
        for (int w = 0; w < 7; ++w) nset += __popc(maskS[tid * MW + w]);
        d_i = (float)nset + (float)(N - nset) * EPSW;
        isq = rsqrtf(d_i);
    }
    float sumd = block_sum((tid < N) ? d_i : 0.f, red, tid);
    if (tid < N) {
        invs[tid] = isq;
        u0[tid]   = sqrtf(d_i) * rsqrtf(sumd);
    }
    __syncthreads();

    // deterministic pseudo-random start vector (register-resident)
    float xi = 0.f;
    if (tid < N) {
        unsigned int h = (unsigned int)tid * 2654435761u + 12345u;
        h ^= h >> 13;
        xi = (float)(h & 0xFFFFu) * (1.0f / 65536.0f) - 0.5f;
    }

    // ---- Phase 2b: deflated power iteration on B = I + S ----
    for (int it = 0; it < NITER; ++it) {
        float zi = (tid < N) ? xi * invs[tid] : 0.f;
        if (tid < N) zv[tid] = zi;
        float total = block_sum(zi, red, tid);      // barrier also publishes zv
        float yi = 0.f;
        if (tid < N) {
            float s = 0.f;
            #pragma unroll
            for (int w = 0; w < 7; ++w) {
                unsigned int mw = maskS[tid * MW + w];
                const int base = w * 32;
                while (mw) {
                    int b = __builtin_ctz(mw);
                    mw &= mw - 1;
                    s += zv[base + b];
                }
            }
            yi = xi + invs[tid] * ((1.0f - EPSW) * s + EPSW * total);
        }
        float proj = block_sum((tid < N) ? yi * u0[tid] : 0.f, red, tid);
        if (tid < N) yi -= proj * u0[tid];
        float nrm = block_sum(yi * yi, red, tid);
        xi = yi * rsqrtf(nrm + 1e-30f);             // ||u||_2 = 1, matches eigh normalization
    }

    // ---- Phase 2c: v = D^{-1/2} u, sign-invariant bipartition, softmax ----
    float vi  = (tid < N) ? xi * invs[tid] : 0.f;
    float avg = block_sum(vi, red, tid) * (1.0f / (float)N);
    bool  pos = vi > 0.f;
    bool  bip = (avg > 0.f) ? pos : !pos;
    float ev  = (tid < N && bip) ? fabsf(vi) : 0.f;

    float mx = block_max((tid < N) ? ev : -1e30f, red, tid);
    float e  = (tid < N) ? expf(ev - mx) : 0.f;
    float se = block_sum(e, red, tid);
    if (tid < N) out[(size_t)p * N + tid] = e / se;
}

extern "C" void kernel_launch(void* const* d_in, const int* in_sizes, int n_in,
                              void* d_out, int out_size, void* d_ws, size_t ws_size,
                              hipStream_t stream) {
    (void)in_sizes; (void)n_in; (void)out_size; (void)d_ws; (void)ws_size;
    const float* qkv = (const float*)d_in[0];
    float* out = (float*)d_out;
    spatial_weighting_kernel<<<dim3(NPROB), dim3(THREADS), 0, stream>>>(qkv, out);
}